// LengthRegulator_68779606278251
// MI455X (gfx1250) — compile-verified
//
#include <hip/hip_runtime.h>
#include <hip/hip_bf16.h>

#ifndef __has_builtin
#define __has_builtin(x) 0
#endif

#if __has_builtin(__builtin_amdgcn_global_load_async_to_lds_b128) && \
    __has_builtin(__builtin_amdgcn_s_wait_asynccnt)
#define LR_USE_ASYNC_LDS 1
#else
#define LR_USE_ASYNC_LDS 0
#endif

// Native vector types (builtins reject HIP_vector_type classes).
typedef int   lr_v4i __attribute__((ext_vector_type(4)));
typedef float lr_v4f __attribute__((ext_vector_type(4)));
typedef __attribute__((address_space(1))) lr_v4i* lr_g4i;   // global int4*
typedef __attribute__((address_space(3))) lr_v4i* lr_l4i;   // LDS int4*

namespace {
constexpr int LR_B = 32;
constexpr int LR_T = 1024;
constexpr int LR_D = 384;
constexpr int LR_TOUT = 4096;
constexpr int LR_D4 = LR_D / 4;          // 96 x (4 floats) per row
constexpr int ROWS_PER_BLOCK = 32;       // 8 waves x 4 rows each
constexpr int GATHER_THREADS = 256;      // 8 wave32s
}

// ---------------------------------------------------------------------------
// Kernel 1: per-batch inclusive scan of max(durations,0) -> ends[b][0..T-1]
// One 1024-thread workgroup per batch; Hillis-Steele scan in LDS.
// ---------------------------------------------------------------------------
__global__ __launch_bounds__(LR_T) void lr_scan_kernel(
    const int* __restrict__ durations, int* __restrict__ ends) {
  __shared__ int s[LR_T];
  const int b = blockIdx.x;
  const int tid = threadIdx.x;
  int v = durations[b * LR_T + tid];
  s[tid] = v < 0 ? 0 : v;
  __syncthreads();
#pragma unroll
  for (int off = 1; off < LR_T; off <<= 1) {
    int t = (tid >= off) ? s[tid - off] : 0;
    __syncthreads();
    s[tid] += t;
    __syncthreads();
  }
  ends[b * LR_T + tid] = s[tid];
}

// ---------------------------------------------------------------------------
// Kernel 2: searchsorted + row gather.
// Grid: (LR_TOUT / ROWS_PER_BLOCK, LR_B). Block: 256 threads (8 wave32s).
// ends[b][:] (4 KB) staged into LDS via CDNA5 async global->LDS load; each
// wave binary-searches per row then copies 384 floats as 3 x float4 / lane,
// with nontemporal (TH=NT) stores for the write-once 201 MB output.
// ---------------------------------------------------------------------------
__global__ __launch_bounds__(GATHER_THREADS) void lr_gather_kernel(
    const float* __restrict__ x, const int* __restrict__ ends,
    const int* __restrict__ target_len, float* __restrict__ out) {
  __shared__ int se[LR_T];
  const int b = blockIdx.y;
  const int tid = threadIdx.x;
  const int wave = tid >> 5;
  const int lane = tid & 31;

  const int* eb = ends + b * LR_T;
#if LR_USE_ASYNC_LDS
  // 256 threads x 16B = 4096B: entire ends[b][:] in one async b128 per lane.
  __builtin_amdgcn_global_load_async_to_lds_b128(
      (lr_g4i)(eb + tid * 4),
      (lr_l4i)(se + tid * 4),
      /*offset=*/0, /*cpol=*/0);
  __builtin_amdgcn_s_wait_asynccnt(0);
#else
  ((lr_v4i*)se)[tid] = ((const lr_v4i*)eb)[tid];
#endif
  __syncthreads();

  const int total = se[LR_T - 1];
  const int tl = target_len[b];
  const int limit = total < tl ? total : tl;

  const lr_v4f* xb = (const lr_v4f*)(x) + (size_t)b * LR_T * LR_D4;
  const int row0 = blockIdx.x * ROWS_PER_BLOCK;

  for (int r = wave; r < ROWS_PER_BLOCK; r += (GATHER_THREADS / 32)) {
    const int pos = row0 + r;
    lr_v4f* orow = (lr_v4f*)(out) + ((size_t)b * LR_TOUT + pos) * LR_D4;
    if (pos < limit) {
      // searchsorted(ends, pos, side='right'): first i with ends[i] > pos.
      int lo = 0, hi = LR_T;
      while (lo < hi) {
        int mid = (lo + hi) >> 1;
        if (se[mid] <= pos) lo = mid + 1; else hi = mid;
      }
      int idx = lo > (LR_T - 1) ? (LR_T - 1) : lo;
      const lr_v4f* xrow = xb + (size_t)idx * LR_D4;
#pragma unroll
      for (int k = 0; k < 3; ++k) {
        lr_v4f v = xrow[lane + k * 32];
        __builtin_nontemporal_store(v, &orow[lane + k * 32]);
      }
    } else {
      lr_v4f z = {0.f, 0.f, 0.f, 0.f};
#pragma unroll
      for (int k = 0; k < 3; ++k) {
        __builtin_nontemporal_store(z, &orow[lane + k * 32]);
      }
    }
  }
}

// ---------------------------------------------------------------------------
// Host entry
// ---------------------------------------------------------------------------
extern "C" void kernel_launch(void* const* d_in, const int* in_sizes, int n_in,
                              void* d_out, int out_size, void* d_ws, size_t ws_size,
                              hipStream_t stream) {
  (void)in_sizes; (void)n_in; (void)out_size; (void)ws_size;
  const float* x        = (const float*)d_in[0];   // (B, T, D) fp32
  const int* durations  = (const int*)d_in[1];     // (B, T) int32
  const int* target_len = (const int*)d_in[2];     // (B,) int32
  float* out            = (float*)d_out;           // (B, T_OUT, D) fp32
  int* ends             = (int*)d_ws;              // B*T int32 = 128 KB scratch

  lr_scan_kernel<<<LR_B, LR_T, 0, stream>>>(durations, ends);

  dim3 grid(LR_TOUT / ROWS_PER_BLOCK, LR_B);
  lr_gather_kernel<<<grid, GATHER_THREADS, 0, stream>>>(x, ends, target_len, out);
}